// SelfCF_21483426415046
// MI455X (gfx1250) — compile-verified
//
#include <hip/hip_runtime.h>
#include <stdint.h>

#define NUM_USER 200000
#define NUM_ITEM 100000
#define NN (NUM_USER + NUM_ITEM)
#define DIM 64
#define BATCH 8192
#define TILE 512                    // edges staged per block per stage
#define BD ((long)BATCH * DIM)
#define EDGE_SALT 0x5EED5EED0000ull
#define USER_SALT 0xA11CE000000000ull
#define ITEM_SALT 0xB0B0000000000ull

// ---------------- RNG: SplitMix64 finalizer -> Bernoulli(0.5) ----------------
__device__ __forceinline__ uint32_t mix32(uint64_t x) {
  x += 0x9E3779B97F4A7C15ull;
  x = (x ^ (x >> 30)) * 0xBF58476D1CE4E5B9ull;
  x = (x ^ (x >> 27)) * 0x94D049BB133111EBull;
  return (uint32_t)(x ^ (x >> 31));
}

// Low 32 bits of a flat shared-memory address == LDS byte offset (HW truncates).
__device__ __forceinline__ uint32_t lds_off(const void* p) {
  return (uint32_t)(uintptr_t)p;
}

// CDNA5 async global->LDS copy (ASYNCcnt-tracked), GV addressing mode.
__device__ __forceinline__ void async_load_b64(uint32_t ldsoff, const void* gaddr) {
  asm volatile("global_load_async_to_lds_b64 %0, %1, off"
               :: "v"(ldsoff), "v"(gaddr) : "memory");
}
#define S_WAIT_ASYNCCNT_3() asm volatile("s_wait_asynccnt 0x3" ::: "memory")

// Native fp32 atomic add at L2 (no-return form), independent of -munsafe-fp-atomics.
__device__ __forceinline__ void atomic_fadd(float* p, float v) {
  asm volatile("global_atomic_add_f32 %0, %1, off" :: "v"(p), "v"(v) : "memory");
}

__device__ __forceinline__ void flush_row(float* __restrict__ y, int row, int l,
                                          const float4& acc) {
  float* p = y + (size_t)row * DIM + l * 4;
  atomic_fadd(p + 0, acc.x);
  atomic_fadd(p + 1, acc.y);
  atomic_fadd(p + 2, acc.z);
  atomic_fadd(p + 3, acc.w);
}

// ---------------- x = concat(user_emb, item_emb) ----------------
__global__ __launch_bounds__(256) void k_init(const float4* __restrict__ ue,
                                              const float4* __restrict__ ie,
                                              float4* __restrict__ x,
                                              long n4u, long n4) {
  long i = (long)blockIdx.x * blockDim.x + threadIdx.x;
  if (i >= n4) return;
  x[i] = (i < n4u) ? ue[i] : ie[i - n4u];
}

__global__ __launch_bounds__(256) void k_zero(float4* __restrict__ p, long n4) {
  long i = (long)blockIdx.x * blockDim.x + threadIdx.x;
  if (i >= n4) return;
  p[i] = make_float4(0.f, 0.f, 0.f, 0.f);
}

// ---------------- SpMM: async-LDS staged edges + run-length atomic reduction ----------------
// adj_rows' first half is sorted (np.unique over u*NUM_ITEM+it), so consecutive edges
// frequently share a row; each 16-lane group walks 32 consecutive edges and register-
// accumulates the float4 partial until the row changes, then flushes one atomic burst.
__global__ __launch_bounds__(256) void k_spmm(
    const int* __restrict__ rows, const int* __restrict__ cols,
    const float* __restrict__ vals, const float* __restrict__ x,
    float* __restrict__ y, long nt, long gstride)
{
  __shared__ int   sr[2][TILE];
  __shared__ int   sc[2][TILE];
  __shared__ float sv[2][TILE];

  const int tid = threadIdx.x;
  const long b0 = blockIdx.x;
  if (b0 >= nt) return;

  const uint32_t offR[2] = { lds_off(&sr[0][0]) + (uint32_t)tid * 8u,
                             lds_off(&sr[1][0]) + (uint32_t)tid * 8u };
  const uint32_t offC[2] = { lds_off(&sc[0][0]) + (uint32_t)tid * 8u,
                             lds_off(&sc[1][0]) + (uint32_t)tid * 8u };
  const uint32_t offV[2] = { lds_off(&sv[0][0]) + (uint32_t)tid * 8u,
                             lds_off(&sv[1][0]) + (uint32_t)tid * 8u };

  auto issue = [&](int b, long t) {
    const long base = t * TILE + (long)tid * 2;   // 2 x 4B elements per thread per array
    async_load_b64(offR[b], rows + base);
    async_load_b64(offC[b], cols + base);
    async_load_b64(offV[b], vals + base);
  };

  issue(0, b0);                 // prologue stage
  int pb = 0;
  const int gid = tid >> 4;     // 16 groups of 16 lanes; group owns 32 consecutive edges
  const int l   = tid & 15;     // lane within the 16-lane row group (4 floats each)

  for (long t = b0; t < nt; t += gstride) {
    const long tn = t + gstride;
    issue(pb ^ 1, (tn < nt) ? tn : t);   // prefetch next tile (dummy reload on last iter)
    S_WAIT_ASYNCCNT_3();                 // 3 newest outstanding -> current tile complete
    __syncthreads();

    int    cur_row = -1;
    float4 acc     = make_float4(0.f, 0.f, 0.f, 0.f);
    #pragma unroll 1
    for (int j = 0; j < 32; ++j) {
      const int el  = (gid << 5) + j;
      const int row = sr[pb][el];
      const int col = sc[pb][el];
      const float v = sv[pb][el];
      if (j < 31)  // warm the path for the next gather row
        __builtin_prefetch(x + (size_t)sc[pb][el + 1] * DIM, 0, 0);
      const uint64_t ge = (uint64_t)t * TILE + (uint64_t)el;
      if (mix32(ge + EDGE_SALT) & 1u) {          // edge dropout keep (p=0.5)
        const float v2 = v * 2.0f;               // 1/(1-EDGE_DROP)
        const float4 xv = *(const float4*)(x + (size_t)col * DIM + l * 4);
        if (row == cur_row) {                    // extend current run in registers
          acc.x += v2 * xv.x; acc.y += v2 * xv.y;
          acc.z += v2 * xv.z; acc.w += v2 * xv.w;
        } else {                                 // row changed: flush previous run
          if (cur_row >= 0) flush_row(y, cur_row, l, acc);
          cur_row = row;
          acc = make_float4(v2 * xv.x, v2 * xv.y, v2 * xv.z, v2 * xv.w);
        }
      }
    }
    if (cur_row >= 0) flush_row(y, cur_row, l, acc);

    __syncthreads();
    pb ^= 1;
  }
}

// Tail edges (< TILE), plain loads so async never reads out of bounds.
__global__ __launch_bounds__(256) void k_spmm_tail(
    const int* __restrict__ rows, const int* __restrict__ cols,
    const float* __restrict__ vals, const float* __restrict__ x,
    float* __restrict__ y, long start, long n)
{
  const int grp = threadIdx.x >> 4;
  const int l   = threadIdx.x & 15;
  for (long e = start + grp; e < n; e += 16) {
    if (mix32((uint64_t)e + EDGE_SALT) & 1u) {
      const int col = cols[e];
      const int row = rows[e];
      const float v2 = vals[e] * 2.0f;
      const float4 xv = *(const float4*)(x + (size_t)col * DIM + l * 4);
      float* p = y + (size_t)row * DIM + l * 4;
      atomic_fadd(p + 0, v2 * xv.x);
      atomic_fadd(p + 1, v2 * xv.y);
      atomic_fadd(p + 2, v2 * xv.z);
      atomic_fadd(p + 3, v2 * xv.w);
    }
  }
}

// ---------------- gather this layer's contribution into d_out (scaled 1/4) ----------------
// d_out layout (float4 units, H = BATCH*16 per section):
//   [0,H)=u_online  [H,2H)=u_target  [2H,3H)=i_online  [3H,4H)=i_target
__global__ __launch_bounds__(256) void k_gather(
    const float4* __restrict__ x, const int* __restrict__ users,
    const int* __restrict__ items, float4* __restrict__ out, int accumulate)
{
  const long H = (long)BATCH * 16;
  const long i = (long)blockIdx.x * blockDim.x + threadIdx.x;
  if (i >= 2 * H) return;
  long dst;
  const float4* src;
  if (i < H) {
    const long r = i >> 4, q = i & 15;
    src = x + ((size_t)users[r] * 16 + q);
    dst = i;
  } else {
    const long j = i - H, r = j >> 4, q = j & 15;
    src = x + (((size_t)NUM_USER + (size_t)items[r]) * 16 + q);
    dst = 2 * H + j;
  }
  const float4 v = *src;
  const float s = 0.25f;          // 1/(N_LAYERS+1)
  float4 r4 = make_float4(v.x * s, v.y * s, v.z * s, v.w * s);
  if (accumulate) {
    const float4 o = out[dst];
    r4.x += o.x; r4.y += o.y; r4.z += o.z; r4.w += o.w;
  }
  out[dst] = r4;
}

// ---------------- target dropout: target = mask ? online*2 : 0 ----------------
__global__ __launch_bounds__(256) void k_targets(float* __restrict__ out) {
  const long i = (long)blockIdx.x * blockDim.x + threadIdx.x;
  if (i >= 2 * BD) return;
  float on; long dst; uint32_t h;
  if (i < BD) {
    on = out[i];            dst = BD + i;             h = mix32((uint64_t)i + USER_SALT);
  } else {
    const long j = i - BD;
    on = out[2 * BD + j];   dst = 3 * BD + j;         h = mix32((uint64_t)j + ITEM_SALT);
  }
  out[dst] = (h >> 31) ? on * 2.0f : 0.0f;   // keep prob 0.5, scale 1/(1-TGT_DROP)
}

// ---------------- host ----------------
extern "C" void kernel_launch(void* const* d_in, const int* in_sizes, int n_in,
                              void* d_out, int out_size, void* d_ws, size_t ws_size,
                              hipStream_t stream)
{
  const float* user_emb = (const float*)d_in[0];
  const float* item_emb = (const float*)d_in[1];
  const float* adj_vals = (const float*)d_in[2];
  const int*   adj_rows = (const int*)d_in[3];
  const int*   adj_cols = (const int*)d_in[4];
  const int*   users    = (const int*)d_in[5];
  const int*   items    = (const int*)d_in[6];
  const long E = in_sizes[2];

  // ws: two ping-pong node-embedding buffers, 76.8 MB each
  float* xa = (float*)d_ws;
  float* xb = xa + (long)NN * DIM;

  const long n4  = (long)NN * DIM / 4;
  const long n4u = (long)NUM_USER * DIM / 4;
  k_init<<<(int)((n4 + 255) / 256), 256, 0, stream>>>(
      (const float4*)user_emb, (const float4*)item_emb, (float4*)xa, n4u, n4);

  const long gthreads = 2L * BATCH * 16;
  k_gather<<<(int)((gthreads + 255) / 256), 256, 0, stream>>>(
      (const float4*)xa, users, items, (float4*)d_out, 0);   // layer-0 term

  const long NT = E / TILE;
  const long grid = (NT < 1024) ? (NT > 0 ? NT : 1) : 1024;
  float* xc = xa;
  float* xn = xb;
  for (int layer = 0; layer < 3; ++layer) {
    k_zero<<<(int)((n4 + 255) / 256), 256, 0, stream>>>((float4*)xn, n4);
    if (NT > 0)
      k_spmm<<<(int)grid, 256, 0, stream>>>(adj_rows, adj_cols, adj_vals, xc, xn, NT, grid);
    if (NT * TILE < E)
      k_spmm_tail<<<1, 256, 0, stream>>>(adj_rows, adj_cols, adj_vals, xc, xn, NT * TILE, E);
    k_gather<<<(int)((gthreads + 255) / 256), 256, 0, stream>>>(
        (const float4*)xn, users, items, (float4*)d_out, 1);
    float* tmp = xc; xc = xn; xn = tmp;
  }
  k_targets<<<(int)((2 * BD + 255) / 256), 256, 0, stream>>>((float*)d_out);
}